// DeformConvV2_18322330485392
// MI455X (gfx1250) — compile-verified
//
#include <hip/hip_runtime.h>
#include <hip/hip_bf16.h>

typedef float v2f __attribute__((ext_vector_type(2)));
typedef float v8f __attribute__((ext_vector_type(8)));

#define PP    9            // K*K taps
#define CC    128
#define HH    64
#define WW    64
#define OCH   256
#define CP    (CC * PP)    // 1152 = GEMM K dim
#define KC    36           // k-chunk = 4 channels x 9 taps
#define NCHUNK (CP / KC)   // 32
#define MT    128          // oc rows per workgroup
#define NT    64           // pixels per workgroup (one image row)
#define KPAD  (KC + 2)     // 38: even (8B-aligned b64), conflict-free stride

__global__ __launch_bounds__(256)
void deform_conv_v2_wmma(const float* __restrict__ im,
                         const float* __restrict__ off,
                         const float* __restrict__ msk,
                         const float* __restrict__ wgt,
                         float* __restrict__ out) {
    // Double-buffered LDS staging (~75 KB of 320 KB/WGP)
    __shared__ __align__(16) float sW[2][MT * KC];        // 128 x 36 weight tiles
    __shared__ __align__(16) float sColT[2][NT][KPAD];    // 64 px x 36 k (transposed)
    __shared__ float sw0[PP * NT], sw1[PP * NT], sw2[PP * NT], sw3[PP * NT];
    __shared__ int   si0[PP * NT], si1[PP * NT], si2[PP * NT], si3[PP * NT];

    const int h    = blockIdx.x;             // image row 0..63
    const int mt   = blockIdx.y;             // oc tile 0..1
    const int b    = blockIdx.z;             // batch 0..7
    const int tid  = threadIdx.x;
    const int lane = tid & 31;
    const int wv   = tid >> 5;               // wave 0..7 -> 16-row M subtile

    // ---- decode sampling geometry once per WG: 9 taps x 64 pixels ----
    for (int e = tid; e < PP * NT; e += 256) {
        const int p  = e >> 6;               // tap 0..8
        const int px = e & 63;               // pixel column
        const float oy = off[(((size_t)b * (2 * PP) + 2 * p    ) * HH + h) * WW + px];
        const float ox = off[(((size_t)b * (2 * PP) + 2 * p + 1) * HH + h) * WW + px];
        const float m  = msk[(((size_t)b * PP + p) * HH + h) * WW + px];
        const float y = (float)(h  - 1 + p / 3) + oy;   // STRIDE=1, PAD=1, DIL=1
        const float x = (float)(px - 1 + p % 3) + ox;
        const float y0f = floorf(y), x0f = floorf(x);
        const int   y0 = (int)y0f,   x0 = (int)x0f;
        const int   y1 = y0 + 1,     x1 = x0 + 1;
        const float wy1 = y - y0f,   wx1 = x - x0f;
        const float wy0 = 1.0f - wy1, wx0 = 1.0f - wx1;
        const bool vy0 = (y0 >= 0) && (y0 < HH), vy1 = (y1 >= 0) && (y1 < HH);
        const bool vx0 = (x0 >= 0) && (x0 < WW), vx1 = (x1 >= 0) && (x1 < WW);
        const int y0c = min(max(y0, 0), HH - 1), y1c = min(max(y1, 0), HH - 1);
        const int x0c = min(max(x0, 0), WW - 1), x1c = min(max(x1, 0), WW - 1);
        sw0[e] = wy0 * wx0 * m * ((vy0 && vx0) ? 1.0f : 0.0f);  si0[e] = y0c * WW + x0c;
        sw1[e] = wy0 * wx1 * m * ((vy0 && vx1) ? 1.0f : 0.0f);  si1[e] = y0c * WW + x1c;
        sw2[e] = wy1 * wx0 * m * ((vy1 && vx0) ? 1.0f : 0.0f);  si2[e] = y1c * WW + x0c;
        sw3[e] = wy1 * wx1 * m * ((vy1 && vx1) ? 1.0f : 0.0f);  si3[e] = y1c * WW + x1c;
    }

    const int    ocb   = mt * MT;
    const float* wbase = wgt + (size_t)ocb * CP;
    const float* imb   = im  + (size_t)b * CC * HH * WW;

    // stage one chunk (weight tile + gathered/transposed col tile) into buffer `buf`
    auto fill = [&](int buf, int t) {
        for (int e = tid; e < MT * KC; e += 256) {
            const int r = e / KC;
            const int k = e - r * KC;
            sW[buf][e] = wbase[(size_t)r * CP + t * KC + k];
        }
        for (int e = tid; e < KC * NT; e += 256) {
            const int row = e >> 6;                  // k within chunk 0..35
            const int px  = e & 63;
            const int c   = t * 4 + row / PP;        // channel
            const int p   = row - (row / PP) * PP;   // tap
            const float* pl = imb + (size_t)c * HH * WW;
            const int s = p * NT + px;
            sColT[buf][px][row] = sw0[s] * pl[si0[s]] + sw1[s] * pl[si1[s]]
                                + sw2[s] * pl[si2[s]] + sw3[s] * pl[si3[s]];
        }
    };

    v8f acc0 = {}, acc1 = {}, acc2 = {}, acc3 = {};
    const int mrow  = wv * 16 + (lane & 15);     // A: both lane halves M=0..15
    const int ncol  = lane & 15;                 // B: N striped across lanes
    const int khalf = (lane >> 4) * 2;           // upper lanes hold K=2,3

    fill(0, 0);
    __syncthreads();

    for (int t = 0; t < NCHUNK; ++t) {
        const int buf = t & 1;
        if (t + 1 < NCHUNK) fill(buf ^ 1, t + 1);  // overlaps with WMMA below

        #pragma unroll
        for (int kk = 0; kk < KC; kk += 4) {
            const v2f a  = *(const v2f*)&sW[buf][mrow * KC + kk + khalf];
            const v2f b0 = *(const v2f*)&sColT[buf][ncol     ][kk + khalf];
            const v2f b1 = *(const v2f*)&sColT[buf][ncol + 16][kk + khalf];
            const v2f b2 = *(const v2f*)&sColT[buf][ncol + 32][kk + khalf];
            const v2f b3 = *(const v2f*)&sColT[buf][ncol + 48][kk + khalf];
            acc0 = __builtin_amdgcn_wmma_f32_16x16x4_f32(false, a, false, b0, (short)0, acc0, false, false);
            acc1 = __builtin_amdgcn_wmma_f32_16x16x4_f32(false, a, false, b1, (short)0, acc1, false, false);
            acc2 = __builtin_amdgcn_wmma_f32_16x16x4_f32(false, a, false, b2, (short)0, acc2, false, false);
            acc3 = __builtin_amdgcn_wmma_f32_16x16x4_f32(false, a, false, b3, (short)0, acc3, false, false);
        }
        __syncthreads();
    }

    // C/D layout: VGPR r -> M=r (lanes 0-15), M=r+8 (lanes 16-31); N = lane&15
    const int Mbase = ocb + wv * 16 + ((lane >> 4) ? 8 : 0);
    const int N     = lane & 15;
    #pragma unroll
    for (int r = 0; r < 8; ++r) {
        const int oc = Mbase + r;
        const size_t o = (((size_t)b * OCH + oc) * HH + h) * WW;
        out[o + N]      = acc0[r];
        out[o + N + 16] = acc1[r];
        out[o + N + 32] = acc2[r];
        out[o + N + 48] = acc3[r];
    }
}

extern "C" void kernel_launch(void* const* d_in, const int* in_sizes, int n_in,
                              void* d_out, int out_size, void* d_ws, size_t ws_size,
                              hipStream_t stream) {
    (void)in_sizes; (void)n_in; (void)out_size; (void)d_ws; (void)ws_size;
    const float* data_im = (const float*)d_in[0];   // [8,128,64,64]
    const float* offset  = (const float*)d_in[1];   // [8,18,64,64]
    const float* mask    = (const float*)d_in[2];   // [8,9,64,64]
    const float* weight  = (const float*)d_in[3];   // [256,1152]
    float* out = (float*)d_out;                     // [8,256,64,64]

    dim3 grid(HH, OCH / MT, 8);   // (64 rows, 2 oc tiles, 8 batches)
    dim3 block(256);
    deform_conv_v2_wmma<<<grid, block, 0, stream>>>(data_im, offset, mask, weight, out);
}